// SimilarityMaxtirx_77000173682787
// MI455X (gfx1250) — compile-verified
//
#include <hip/hip_runtime.h>

typedef __attribute__((ext_vector_type(2))) float v2f;
typedef __attribute__((ext_vector_type(8))) float v8f;

#define B_ 32
#define N_ 400
#define M_ 50
#define D_ 128
#define VERY_NEG (-1e30f)

// ---------------------------------------------------------------------------
// Kernel 1: part_c[b,n] = c[b,n,:] . W[0:128],  part_q[b,m] = q[b,m,:] . W[128:256]
// One wave32 per row; lane-strided loads + shuffle reduction.
// ---------------------------------------------------------------------------
__global__ __launch_bounds__(256) void parts_kernel(
    const float* __restrict__ c, const float* __restrict__ q,
    const float* __restrict__ W,
    float* __restrict__ part_c, float* __restrict__ part_q)
{
    int gid  = blockIdx.x * blockDim.x + threadIdx.x;
    int wave = gid >> 5;
    int lane = gid & 31;
    const int totalC = B_ * N_;   // 12800
    const int totalQ = B_ * M_;   // 1600
    if (wave >= totalC + totalQ) return;

    const float* row;
    const float* w;
    float* dst;
    if (wave < totalC) {
        row = c + (size_t)wave * D_;
        w   = W;                 // w_c
        dst = part_c + wave;
    } else {
        int r = wave - totalC;
        row = q + (size_t)r * D_;
        w   = W + D_;            // w_q
        dst = part_q + r;
    }

    float s = 0.0f;
#pragma unroll
    for (int k = 0; k < D_ / 32; ++k)
        s += row[lane + 32 * k] * w[lane + 32 * k];

#pragma unroll
    for (int off = 16; off >= 1; off >>= 1)
        s += __shfl_xor(s, off, 32);

    if (lane == 0) *dst = s;
}

// ---------------------------------------------------------------------------
// Kernel 2: one wave per (b, n-tile). Computes a 16(n) x 64(m) output strip
// via 4 independent V_WMMA_F32_16X16X4_F32 accumulator chains over K=128.
//
// A(16x4 f32) layout: lane L -> row n0 + (L%16), VGPR pair = K = kk + 2*(L/16) + {0,1}
// B(4x16 f32) layout: lane L -> col m  = (L%16), VGPR pair = K = kk + 2*(L/16) + {0,1}
// C/D (16x16 f32):    VGPR v, lane L -> row v + 8*(L/16), col (L%16)
// ---------------------------------------------------------------------------
__global__ __launch_bounds__(32) void sim_wmma_kernel(
    const float* __restrict__ c, const float* __restrict__ q,
    const int* __restrict__ cmask, const int* __restrict__ qmask,
    const float* __restrict__ W,
    const float* __restrict__ part_c, const float* __restrict__ part_q,
    float* __restrict__ out)
{
    const int nt   = blockIdx.x;          // 0..24
    const int b    = blockIdx.y;          // 0..31
    const int lane = threadIdx.x;         // 0..31
    const int half = lane >> 4;           // lane group (K sub-pair select)
    const int l16  = lane & 15;
    const int n0   = nt * 16;

    // Per-lane A base: c row (n0 + l16), starting at K offset 2*half
    const float* crow = c + (((size_t)b * N_ + (n0 + l16)) * D_) + 2 * half;
    const float* wcq  = W + 2 * D_ + 2 * half;   // w_cq with same K pattern

    // Per-lane B bases for the 4 m-tiles (clamped so loads stay in-bounds;
    // stores are guarded below, EXEC stays all-ones for WMMA)
    int mc[4];
#pragma unroll
    for (int t = 0; t < 4; ++t) {
        int m = t * 16 + l16;
        mc[t] = (m < M_) ? m : (M_ - 1);
    }
    const float* qr0 = q + (((size_t)b * M_ + mc[0]) * D_) + 2 * half;
    const float* qr1 = q + (((size_t)b * M_ + mc[1]) * D_) + 2 * half;
    const float* qr2 = q + (((size_t)b * M_ + mc[2]) * D_) + 2 * half;
    const float* qr3 = q + (((size_t)b * M_ + mc[3]) * D_) + 2 * half;

    v8f acc0 = {}, acc1 = {}, acc2 = {}, acc3 = {};

#pragma unroll 8
    for (int kk = 0; kk < D_; kk += 4) {
        v2f a = *(const v2f*)(crow + kk);
        v2f w = *(const v2f*)(wcq + kk);
        a.x *= w.x;                 // fold w_cq into A
        a.y *= w.y;
        v2f b0 = *(const v2f*)(qr0 + kk);
        v2f b1 = *(const v2f*)(qr1 + kk);
        v2f b2 = *(const v2f*)(qr2 + kk);
        v2f b3 = *(const v2f*)(qr3 + kk);

        acc0 = __builtin_amdgcn_wmma_f32_16x16x4_f32(false, a, false, b0, (short)0, acc0, false, false);
        acc1 = __builtin_amdgcn_wmma_f32_16x16x4_f32(false, a, false, b1, (short)0, acc1, false, false);
        acc2 = __builtin_amdgcn_wmma_f32_16x16x4_f32(false, a, false, b2, (short)0, acc2, false, false);
        acc3 = __builtin_amdgcn_wmma_f32_16x16x4_f32(false, a, false, b3, (short)0, acc3, false, false);
    }

    // Column-side terms (my column within each m-tile)
    float pq[4];
    int   qm[4];
#pragma unroll
    for (int t = 0; t < 4; ++t) {
        int m   = t * 16 + l16;
        bool ok = (m < M_);
        pq[t] = ok ? part_q[b * M_ + m] : 0.0f;
        qm[t] = ok ? qmask[b * M_ + m] : 0;
    }

    v8f accs[4] = {acc0, acc1, acc2, acc3};

#pragma unroll
    for (int v = 0; v < 8; ++v) {
        int    n  = n0 + v + 8 * half;
        float  pc = part_c[b * N_ + n];
        int    cm = cmask[b * N_ + n];
        size_t rowbase = ((size_t)b * N_ + n) * M_;
#pragma unroll
        for (int t = 0; t < 4; ++t) {
            int m = t * 16 + l16;
            if (m < M_) {
                float val = accs[t][v] + pc + pq[t];
                if (!(cm && qm[t])) val += VERY_NEG;
                out[rowbase + m] = val;
            }
        }
    }
}

// ---------------------------------------------------------------------------
extern "C" void kernel_launch(void* const* d_in, const int* in_sizes, int n_in,
                              void* d_out, int out_size, void* d_ws, size_t ws_size,
                              hipStream_t stream)
{
    const float* c     = (const float*)d_in[0];
    const float* q     = (const float*)d_in[1];
    const int*   cmask = (const int*)d_in[2];
    const int*   qmask = (const int*)d_in[3];
    const float* W     = (const float*)d_in[4];
    float*       out   = (float*)d_out;

    float* part_c = (float*)d_ws;          // B*N floats
    float* part_q = part_c + B_ * N_;      // B*M floats

    // Kernel 1: 14400 waves, 8 waves (256 threads) per block
    int waves   = B_ * N_ + B_ * M_;
    int threads = 256;
    int blocks  = (waves * 32 + threads - 1) / threads;
    parts_kernel<<<blocks, threads, 0, stream>>>(c, q, W, part_c, part_q);

    // Kernel 2: one wave per (n-tile, batch)
    dim3 grid(N_ / 16, B_);
    sim_wmma_kernel<<<grid, 32, 0, stream>>>(c, q, cmask, qmask, W,
                                             part_c, part_q, out);
}